// Topo_12300786336214
// MI455X (gfx1250) — compile-verified
//
#include <hip/hip_runtime.h>
#include <hip/hip_bf16.h>

typedef __attribute__((ext_vector_type(2))) float v2f;
typedef __attribute__((ext_vector_type(8))) float v8f;

#define LS 33               // LDS row stride (gcd(33,64)=1 -> conflict-free columns)
#define REEIG_EPS 1e-4f
#define NSWEEP 8

// ---------------------------------------------------------------------------
// Kernel 1: per-sample n: xm = mean_T(x); d = xm - mean_C(xm);
//           cov = d^T d / 255 + 1e-8 I  via V_WMMA_F32_16X16X4_F32
// block = 256 threads (8 waves), grid = 256 (one per sample)
// ---------------------------------------------------------------------------
__global__ __launch_bounds__(256) void k_cov(const float* __restrict__ x,
                                             float* __restrict__ cov)
{
    __shared__ float d[256 * LS];     // d[c][v], v padded with zeros to 32
    __shared__ float part[8 * LS];
    __shared__ float mu[LS];

    const int tid  = threadIdx.x;
    const int lane = tid & 31;
    const int w    = tid >> 5;
    const int n    = blockIdx.x;

    // zero LDS (padding columns v=25..31 must stay 0 for the WMMA tiles)
    for (int i = tid; i < 256 * LS; i += 256) d[i] = 0.0f;
    __syncthreads();

    // mean over T: wave w handles channels c = w, w+8, ...
    const size_t xb = (size_t)n * 256 * 64 * 25;
    for (int c = w; c < 256; c += 8) {
        if (lane < 25) {
            const float* p = x + xb + (size_t)c * 64 * 25 + lane;
            float acc = 0.0f;
            #pragma unroll 8
            for (int t = 0; t < 64; ++t) acc += p[t * 25];
            d[c * LS + lane] = acc * (1.0f / 64.0f);
        }
    }
    __syncthreads();

    // channel mean (8 wave-partials, then combine)
    if (lane < 25) {
        float s = 0.0f;
        for (int c = w * 32; c < w * 32 + 32; ++c) s += d[c * LS + lane];
        part[w * LS + lane] = s;
    }
    __syncthreads();
    if (w == 0 && lane < 25) {
        float s = 0.0f;
        for (int g = 0; g < 8; ++g) s += part[g * LS + lane];
        mu[lane] = s * (1.0f / 256.0f);
    }
    __syncthreads();

    // subtract channel mean: thread t owns channel c = t
    for (int v = 0; v < 25; ++v) d[tid * LS + v] -= mu[v];
    __syncthreads();

    // cov tile GEMM: 32x32 padded output, 4 tiles of 16x16, K=256 in steps of 4
    if (w < 4) {
        const int tm   = (w >> 1) * 16;
        const int tn   = (w & 1) * 16;
        const int row  = lane & 15;          // M (for A) / N (for B) within tile
        const int koff = (lane >> 4) << 1;   // K sub-offset per lane half
        v8f acc = {0.f, 0.f, 0.f, 0.f, 0.f, 0.f, 0.f, 0.f};
        for (int k0 = 0; k0 < 256; k0 += 4) {
            const int ka = (k0 + koff) * LS;
            v2f a, b;
            a.x = d[ka      + tm + row];     // A[m][k]   = d[k][m]  (A = d^T)
            a.y = d[ka + LS + tm + row];
            b.x = d[ka      + tn + row];     // B[k][n]   = d[k][n]
            b.y = d[ka + LS + tn + row];
            acc = __builtin_amdgcn_wmma_f32_16x16x4_f32(
                      false, a, false, b, (short)0, acc, false, false);
        }
        const int N     = tn + row;
        const int mbase = tm + ((lane >> 4) << 3);
        if (N < 25) {
            #pragma unroll
            for (int r = 0; r < 8; ++r) {
                const int M = mbase + r;
                if (M < 25) {
                    float v = acc[r] * (1.0f / 255.0f);
                    if (M == N) v += 1e-8f;
                    cov[(size_t)n * 625 + M * 25 + N] = v;
                }
            }
        }
    }
}

// ---------------------------------------------------------------------------
// Kernel 2 helpers: one wave (32 lanes) per sample; matrices in LDS, stride 33.
// ---------------------------------------------------------------------------

// D = W^T * S * W ; S is pin x pin, W is pin x pout (row-major), T1 scratch.
// D may alias S, must not alias T1.
__device__ static inline void bimap(const float* S, const float* W, float* T1,
                                    float* D, int pin, int pout, int l)
{
    if (l < pout) {
        for (int i = 0; i < pin; ++i) {
            float s = 0.0f;
            for (int k = 0; k < pin; ++k) s += S[i * LS + k] * W[k * pout + l];
            T1[i * LS + l] = s;
        }
    }
    __syncthreads();
    if (l < pout) {
        for (int i = 0; i < pout; ++i) {
            float s = 0.0f;
            for (int k = 0; k < pin; ++k) s += W[k * pout + i] * T1[k * LS + l];
            D[i * LS + l] = s;
        }
    }
    __syncthreads();
}

// Parallel-ordered (round-robin) two-sided Jacobi: A -> ~diag, U accumulates
// eigenvectors (A0 = U diag(A) U^T). All floor(P/2) disjoint rotations of a
// round are computed from one snapshot, then rows then columns are updated
// with no intra-phase barriers (disjoint index pairs).
__device__ static inline void jacobi(float* A, float* U, int p, int l,
                                     int* PP, int* QQ, float* CC, float* SS)
{
    if (l < p)
        for (int i = 0; i < p; ++i) U[i * LS + l] = (i == l) ? 1.0f : 0.0f;
    __syncthreads();

    const int P    = (p & 1) ? p + 1 : p;   // add dummy player if p odd
    const int half = P >> 1;

    for (int sw = 0; sw < NSWEEP; ++sw) {
        for (int r = 0; r < P - 1; ++r) {
            // phase 0: rotation parameters for each pair of this round
            if (l < half) {
                int a = (l == 0) ? 0 : 1 + ((l - 1 + r) % (P - 1));
                int b = 1 + ((P - 2 - l + r) % (P - 1));
                int pi = (a < b) ? a : b;
                int qi = (a < b) ? b : a;
                float c = 1.0f, s = 0.0f;
                if (qi < p) {
                    float apq = A[pi * LS + qi];
                    if (fabsf(apq) > 1e-28f) {
                        float app = A[pi * LS + pi];
                        float aqq = A[qi * LS + qi];
                        float tau = (aqq - app) / (2.0f * apq);
                        float t = ((tau >= 0.0f) ? 1.0f : -1.0f) /
                                  (fabsf(tau) + sqrtf(1.0f + tau * tau));
                        c = 1.0f / sqrtf(1.0f + t * t);
                        s = t * c;
                    }
                } else {                     // dummy pair -> identity rotation
                    pi = 0; qi = 1;
                }
                PP[l] = pi; QQ[l] = qi; CC[l] = c; SS[l] = s;
            }
            __syncthreads();
            // phase 1: row updates (disjoint rows across pairs)
            if (l < p) {
                for (int i = 0; i < half; ++i) {
                    int pi = PP[i], qi = QQ[i];
                    float c = CC[i], s = SS[i];
                    float a0 = A[pi * LS + l], a1 = A[qi * LS + l];
                    A[pi * LS + l] = c * a0 - s * a1;
                    A[qi * LS + l] = s * a0 + c * a1;
                }
            }
            __syncthreads();
            // phase 2: column updates + eigenvector accumulation
            if (l < p) {
                for (int i = 0; i < half; ++i) {
                    int pi = PP[i], qi = QQ[i];
                    float c = CC[i], s = SS[i];
                    float a0 = A[l * LS + pi], a1 = A[l * LS + qi];
                    A[l * LS + pi] = c * a0 - s * a1;
                    A[l * LS + qi] = s * a0 + c * a1;
                    float u0 = U[l * LS + pi], u1 = U[l * LS + qi];
                    U[l * LS + pi] = c * u0 - s * u1;
                    U[l * LS + qi] = s * u0 + c * u1;
                }
            }
            __syncthreads();
        }
    }
}

// out = U * f(diag(A)) * U^T ; f = max(.,eps) or log(max(.,eps))
__device__ static inline void recon(const float* A, const float* U, float* out,
                                    int p, int l, bool dolog)
{
    if (l < p) {
        for (int i = 0; i < p; ++i) {
            float s = 0.0f;
            for (int k = 0; k < p; ++k) {
                float wk = fmaxf(A[k * LS + k], REEIG_EPS);
                if (dolog) wk = logf(wk);
                s += U[i * LS + k] * wk * U[l * LS + k];
            }
            out[i * LS + l] = s;
        }
    }
    __syncthreads();
}

// ---------------------------------------------------------------------------
// Kernel 2: BiMap/ReEig chain + fused ReEig+LogEig -> feat (256 x 100)
// block = 32 threads (one wave), grid = 256
// ---------------------------------------------------------------------------
__global__ __launch_bounds__(32) void k_spd(const float* __restrict__ cov,
                                            const float* __restrict__ W1,
                                            const float* __restrict__ W2,
                                            const float* __restrict__ W3,
                                            float* __restrict__ feat)
{
    __shared__ float A[32 * LS], U[32 * LS], T[32 * LS];
    __shared__ int   PP[16], QQ[16];
    __shared__ float CC[16], SS[16];

    const int l = threadIdx.x;
    const int n = blockIdx.x;

    for (int idx = l; idx < 625; idx += 32)
        A[(idx / 25) * LS + (idx % 25)] = cov[(size_t)n * 625 + idx];
    __syncthreads();

    bimap(A, W1, T, A, 25, 20, l);            // A = W1^T cov W1   (20x20)
    jacobi(A, U, 20, l, PP, QQ, CC, SS);
    recon(A, U, T, 20, l, false);             // T = reeig         (20x20)

    bimap(T, W2, U, A, 20, 15, l);            // A = W2^T T W2     (15x15)
    jacobi(A, U, 15, l, PP, QQ, CC, SS);
    recon(A, U, T, 15, l, false);             // T = reeig         (15x15)

    bimap(T, W3, U, A, 15, 10, l);            // A = W3^T T W3     (10x10)
    jacobi(A, U, 10, l, PP, QQ, CC, SS);
    recon(A, U, T, 10, l, true);              // T = logeig(reeig) (10x10)

    for (int idx = l; idx < 100; idx += 32)
        feat[n * 100 + idx] = T[(idx / 10) * LS + (idx % 10)];
}

// ---------------------------------------------------------------------------
// Kernel 3: out = feat(256x100) @ fc_w^T(100x64) + fc_b  via f32 WMMA
// grid = 16 M-tiles, block = 128 (4 waves = 4 N-tiles), K = 100 = 25 steps
// ---------------------------------------------------------------------------
__global__ __launch_bounds__(128) void k_fc(const float* __restrict__ feat,
                                            const float* __restrict__ fcw,
                                            const float* __restrict__ fcb,
                                            float* __restrict__ out)
{
    const int lane = threadIdx.x & 31;
    const int wv   = threadIdx.x >> 5;   // N tile 0..3
    const int mt   = blockIdx.x;         // M tile 0..15
    const int row  = lane & 15;
    const int koff = (lane >> 4) << 1;
    const int m    = mt * 16 + row;      // A row index
    const int nn   = wv * 16 + row;      // B column index (fc output unit)

    v8f acc = {0.f, 0.f, 0.f, 0.f, 0.f, 0.f, 0.f, 0.f};
    for (int k0 = 0; k0 < 100; k0 += 4) {
        v2f a, b;
        a.x = feat[m * 100 + k0 + koff];
        a.y = feat[m * 100 + k0 + koff + 1];
        b.x = fcw[nn * 100 + k0 + koff];      // B[k][n] = fc_w[n][k]
        b.y = fcw[nn * 100 + k0 + koff + 1];
        acc = __builtin_amdgcn_wmma_f32_16x16x4_f32(
                  false, a, false, b, (short)0, acc, false, false);
    }

    const int mbase = mt * 16 + ((lane >> 4) << 3);
    const int N     = wv * 16 + row;
    const float bias = fcb[N];
    #pragma unroll
    for (int r = 0; r < 8; ++r)
        out[(mbase + r) * 64 + N] = acc[r] + bias;
}

// ---------------------------------------------------------------------------
extern "C" void kernel_launch(void* const* d_in, const int* in_sizes, int n_in,
                              void* d_out, int out_size, void* d_ws, size_t ws_size,
                              hipStream_t stream)
{
    (void)in_sizes; (void)n_in; (void)out_size; (void)ws_size;
    const float* x   = (const float*)d_in[0];   // (256,256,64,25)
    const float* W1  = (const float*)d_in[1];   // (1,25,20)
    const float* W2  = (const float*)d_in[2];   // (1,20,15)
    const float* W3  = (const float*)d_in[3];   // (1,15,10)
    const float* fcw = (const float*)d_in[4];   // (64,100)
    const float* fcb = (const float*)d_in[5];   // (64,)
    float* out = (float*)d_out;                 // (256,64) f32

    float* cov  = (float*)d_ws;                 // 256*625 f32 = 640 KB
    float* feat = cov + 256 * 625;              // 256*100 f32 = 100 KB

    k_cov<<<256, 256, 0, stream>>>(x, cov);
    k_spd<<<256,  32, 0, stream>>>(cov, W1, W2, W3, feat);
    k_fc <<< 16, 128, 0, stream>>>(feat, fcw, fcb, out);
}